// GNN_69861938036792
// MI455X (gfx1250) — compile-verified
//
#include <hip/hip_runtime.h>
#include <hip/hip_bf16.h>

// ---------------------------------------------------------------------------
// 3-layer GCN (PyG GCNConv semantics) + mean-pool + FC for MI455X (gfx1250).
// GEMMs use V_WMMA_F32_16X16X4_F32 (fp32 WMMA -> exact precision; GEMM is
// only ~2.8 GFLOP total, the edge gather/scatter is the bound, and it lives
// in the 192MB L2). Message passing is wave-per-edge, fully coalesced.
// ReLU is a compile-time template parameter so layer-1 GEMM carries zero
// activation overhead and layers 2/3 pay one v_max per element.
// ---------------------------------------------------------------------------

#define N_NODES   50000
#define N_EDGES   800000
#define IN_DIM    96
#define HID       96
#define OUT_DIM   32
#define N_GRAPHS  256

typedef __attribute__((ext_vector_type(2))) float v2f;
typedef __attribute__((ext_vector_type(8))) float v8f;

// ---------------------------------------------------------------------------
// Degree / normalization
// ---------------------------------------------------------------------------
__global__ void k_deg_init(float* __restrict__ deg) {
    int i = blockIdx.x * blockDim.x + threadIdx.x;
    if (i < N_NODES) deg[i] = 1.0f;            // self-loop contribution
}

__global__ void k_deg_edges(const int* __restrict__ ei, float* __restrict__ deg) {
    int e = blockIdx.x * blockDim.x + threadIdx.x;
    if (e < N_EDGES) atomicAdd(&deg[ei[N_EDGES + e]], 1.0f);   // col = targets
}

__global__ void k_deg_rsqrt(float* __restrict__ deg) {
    int i = blockIdx.x * blockDim.x + threadIdx.x;
    if (i < N_NODES) deg[i] = rsqrtf(deg[i]);  // deg >= 1 always (self loops)
}

// ---------------------------------------------------------------------------
// C[M x 96] = A[M x 96] @ B[96 x 96] via V_WMMA_F32_16X16X4_F32.
// Block = 128 threads = 4 waves. Each wave owns one 16x16 C tile.
// blockIdx.y selects the 16-wide N tile; its B slice (96x16) is staged in LDS.
// RELU (previous layer's activation) fused into the A load at compile time.
//
// Fragment layouts (CDNA5 ISA):
//   A 16x4 f32  -> v2f; lanes 0-15: K={kk,kk+1}, lanes 16-31: K={kk+2,kk+3}
//   B  4x16 f32 -> v2f; same K split, N = lane&15
//   C 16x16 f32 -> v8f; VGPR r: lanes 0-15 (M=r), lanes 16-31 (M=r+8)
// ---------------------------------------------------------------------------
template <bool RELU>
__global__ __launch_bounds__(128)
void k_gemm_wmma(const float* __restrict__ A, const float* __restrict__ B,
                 float* __restrict__ C) {
    __shared__ float Bs[96 * 16];

    const int tn = blockIdx.y * 16;
    // Cooperative load of B[:, tn:tn+16] into LDS (1536 floats / 128 threads).
    for (int t = threadIdx.x; t < 96 * 16; t += 128) {
        int k = t >> 4, n = t & 15;
        Bs[t] = B[k * HID + tn + n];
    }
    __syncthreads();

    const int wave = threadIdx.x >> 5;
    const int lane = threadIdx.x & 31;
    const int tile = blockIdx.x * 4 + wave;     // 16-row M tile index
    if (tile >= (N_NODES / 16)) return;         // wave-uniform: EXEC stays all-1s

    const int mlo = lane & 15;                  // M within tile (A) / N within tile (B,C)
    const int hi  = lane >> 4;                  // 0 -> K pair {0,1}, 1 -> K pair {2,3}
    const float* __restrict__ arow = A + (size_t)(tile * 16 + mlo) * IN_DIM;

    v8f acc = {};
#pragma unroll
    for (int kk = 0; kk < 96; kk += 4) {
        const int k0 = kk + 2 * hi;
        float ax = arow[k0];
        float ay = arow[k0 + 1];
        if (RELU) {
            ax = (ax > 0.0f) ? ax : 0.0f;       // single v_max, no cndmask
            ay = (ay > 0.0f) ? ay : 0.0f;
        }
        v2f a = { ax, ay };
        v2f b = { Bs[k0 * 16 + mlo], Bs[(k0 + 1) * 16 + mlo] };
        // (neg_a, A, neg_b, B, c_mod, C, reuse_a, reuse_b)
        acc = __builtin_amdgcn_wmma_f32_16x16x4_f32(
            false, a, false, b, (short)0, acc, false, false);
    }

    const int n     = tn + mlo;
    const int mbase = tile * 16 + hi * 8;
#pragma unroll
    for (int r = 0; r < 8; ++r)
        C[(size_t)(mbase + r) * HID + n] = acc[r];
}

// ---------------------------------------------------------------------------
// agg[i] = dis[i]^2 * hW[i] + bias   (self-loop term + bias, init of scatter dst)
// ---------------------------------------------------------------------------
__global__ void k_selfloop_bias(const float* __restrict__ hW,
                                const float* __restrict__ dis,
                                const float* __restrict__ bias,
                                float* __restrict__ agg) {
    int idx = blockIdx.x * blockDim.x + threadIdx.x;
    if (idx >= N_NODES * HID) return;
    int i = idx / HID, f = idx - i * HID;
    float d = dis[i];
    agg[idx] = d * d * hW[idx] + bias[f];
}

// ---------------------------------------------------------------------------
// Wave-per-edge scatter: agg[col] += dis[row]*dis[col] * hW[row]
// 32 lanes cover the 96 features (3 coalesced floats each) -> coalesced
// 384B gathers + coalesced global_atomic_add_f32, all L2-resident.
// ---------------------------------------------------------------------------
__global__ __launch_bounds__(256)
void k_edge_scatter(const int* __restrict__ ei, const float* __restrict__ hW,
                    const float* __restrict__ dis, float* __restrict__ agg) {
    int warp = (blockIdx.x * blockDim.x + threadIdx.x) >> 5;
    int lane = threadIdx.x & 31;
    if (warp >= N_EDGES) return;
    int r = ei[warp];               // source
    int c = ei[N_EDGES + warp];     // target
    float nrm = dis[r] * dis[c];
    const float* __restrict__ src = hW  + (size_t)r * HID;
    float*       __restrict__ dst = agg + (size_t)c * HID;
#pragma unroll
    for (int j = 0; j < 3; ++j) {
        int f = lane + 32 * j;
        atomicAdd(&dst[f], nrm * src[f]);
    }
}

// ---------------------------------------------------------------------------
// Pooling + FC
// ---------------------------------------------------------------------------
__global__ void k_zero(float* __restrict__ p, int n) {
    int i = blockIdx.x * blockDim.x + threadIdx.x;
    if (i < n) p[i] = 0.0f;
}

__global__ __launch_bounds__(256)
void k_pool(const int* __restrict__ batch, const float* __restrict__ h,
            float* __restrict__ sums, float* __restrict__ cnt) {
    int warp = (blockIdx.x * blockDim.x + threadIdx.x) >> 5;
    int lane = threadIdx.x & 31;
    if (warp >= N_NODES) return;
    int g = batch[warp];
    const float* __restrict__ src = h + (size_t)warp * HID;
    float* __restrict__ dst = sums + (size_t)g * HID;
#pragma unroll
    for (int j = 0; j < 3; ++j) {
        int f = lane + 32 * j;
        float v = src[f];
        atomicAdd(&dst[f], (v > 0.0f) ? v : 0.0f);   // ReLU fused here
    }
    if (lane == 0) atomicAdd(&cnt[g], 1.0f);
}

__global__ void k_fc(const float* __restrict__ sums, const float* __restrict__ cnt,
                     const float* __restrict__ fc_w, const float* __restrict__ fc_b,
                     float* __restrict__ out) {
    int idx = blockIdx.x * blockDim.x + threadIdx.x;
    if (idx >= N_GRAPHS * OUT_DIM) return;
    int g = idx / OUT_DIM, o = idx - g * OUT_DIM;
    float inv = 1.0f / fmaxf(cnt[g], 1.0f);
    float acc = fc_b[o];
#pragma unroll 8
    for (int f = 0; f < HID; ++f)
        acc = fmaf(sums[g * HID + f] * inv, fc_w[f * OUT_DIM + o], acc);
    out[idx] = acc;
}

// ---------------------------------------------------------------------------
// Launch
// ---------------------------------------------------------------------------
extern "C" void kernel_launch(void* const* d_in, const int* in_sizes, int n_in,
                              void* d_out, int out_size, void* d_ws, size_t ws_size,
                              hipStream_t stream) {
    (void)in_sizes; (void)n_in; (void)out_size; (void)ws_size;

    const float* x     = (const float*)d_in[0];
    const int*   ei    = (const int*)  d_in[1];   // [2, E] flat: [0..E)=row, [E..2E)=col
    const int*   batch = (const int*)  d_in[2];
    const float* W1    = (const float*)d_in[3];
    const float* b1    = (const float*)d_in[4];
    const float* W2    = (const float*)d_in[5];
    const float* b2    = (const float*)d_in[6];
    const float* W3    = (const float*)d_in[7];
    const float* b3    = (const float*)d_in[8];
    const float* fc_w  = (const float*)d_in[9];
    const float* fc_b  = (const float*)d_in[10];
    float* out = (float*)d_out;

    // Workspace partition (floats): dis | bufA | bufB | bufC | sums | cnt
    float* dis  = (float*)d_ws;
    float* bufA = dis  + N_NODES;
    float* bufB = bufA + (size_t)N_NODES * HID;
    float* bufC = bufB + (size_t)N_NODES * HID;
    float* sums = bufC + (size_t)N_NODES * HID;
    float* cnt  = sums + (size_t)N_GRAPHS * HID;

    const int T = 256;
    dim3 gemm_grid((N_NODES / 16 + 3) / 4, HID / 16);   // (782, 6), 4 waves/block

    // --- normalization -----------------------------------------------------
    k_deg_init  <<<(N_NODES + T - 1) / T, T, 0, stream>>>(dis);
    k_deg_edges <<<(N_EDGES + T - 1) / T, T, 0, stream>>>(ei, dis);
    k_deg_rsqrt <<<(N_NODES + T - 1) / T, T, 0, stream>>>(dis);

    const int nodeF   = N_NODES * HID;
    const int edgeWps = (N_EDGES * 32 + T - 1) / T;     // wave-per-edge
    const int nodeWps = (N_NODES * 32 + T - 1) / T;     // wave-per-node

    // --- layer 1: x -> bufA ------------------------------------------------
    k_gemm_wmma<false><<<gemm_grid, 128, 0, stream>>>(x, W1, bufB);
    k_selfloop_bias<<<(nodeF + T - 1) / T, T, 0, stream>>>(bufB, dis, b1, bufA);
    k_edge_scatter <<<edgeWps, T, 0, stream>>>(ei, bufB, dis, bufA);

    // --- layer 2: relu(bufA) -> bufC ---------------------------------------
    k_gemm_wmma<true><<<gemm_grid, 128, 0, stream>>>(bufA, W2, bufB);
    k_selfloop_bias<<<(nodeF + T - 1) / T, T, 0, stream>>>(bufB, dis, b2, bufC);
    k_edge_scatter <<<edgeWps, T, 0, stream>>>(ei, bufB, dis, bufC);

    // --- layer 3: relu(bufC) -> bufA ---------------------------------------
    k_gemm_wmma<true><<<gemm_grid, 128, 0, stream>>>(bufC, W3, bufB);
    k_selfloop_bias<<<(nodeF + T - 1) / T, T, 0, stream>>>(bufB, dis, b3, bufA);
    k_edge_scatter <<<edgeWps, T, 0, stream>>>(ei, bufB, dis, bufA);

    // --- mean pool (ReLU fused) + FC ---------------------------------------
    k_zero<<<(N_GRAPHS * HID + N_GRAPHS + T - 1) / T, T, 0, stream>>>(
        sums, N_GRAPHS * HID + N_GRAPHS);                // sums and cnt contiguous
    k_pool<<<nodeWps, T, 0, stream>>>(batch, bufA, sums, cnt);
    k_fc  <<<(N_GRAPHS * OUT_DIM + T - 1) / T, T, 0, stream>>>(sums, cnt, fc_w, fc_b, out);
}